// ImportanceHDRiLight_71880572666338
// MI455X (gfx1250) — compile-verified
//
#include <hip/hip_runtime.h>
#include <hip/hip_bf16.h>
#include <math.h>

typedef __attribute__((ext_vector_type(2))) float v2f;
typedef __attribute__((ext_vector_type(8))) float v8f;

#define HDRI_H 256
#define HDRI_W 512
#define HDRI_HW (HDRI_H * HDRI_W)   // 131072
#define NPR 512
#define NSR 8
#define CHUNK 1024                  // dirs per chunk (one wave's job)
#define NCHUNK (HDRI_HW / CHUNK)    // 128
#define NSEG 16                     // segments per ray-tile (8 chunks each)
#define RAYS_PER_WG 16

// ---------------------------------------------------------------------------
// Kernel 0: pre-swizzle dirs + luminance into the WMMA B-matrix lane layout.
// For tile T (16 dirs), 32 float4 records:
//   record[T*32 + n]      = (dir.x, dir.y, intens, 0)   -> lanes 0..15 (K0,K1)
//   record[T*32 + 16 + n] = (dir.z, 0,     intens, 0)   -> lanes 16..31 (K2,K3)
// ---------------------------------------------------------------------------
__global__ void hdri_pack_kernel(const float* __restrict__ envs,
                                 const float* __restrict__ dirs,
                                 float4* __restrict__ packed) {
    int i = blockIdx.x * blockDim.x + threadIdx.x;
    if (i < HDRI_HW) {
        const float* e = envs + 3 * i;
        const float I = sqrtf(e[0] * e[0] + e[1] * e[1] + e[2] * e[2]);
        const float* d = dirs + 3 * i;
        const int tile = i >> 4;
        const int slot = i & 15;
        packed[tile * 32 + slot]      = make_float4(d[0], d[1], I, 0.0f);
        packed[tile * 32 + 16 + slot] = make_float4(d[2], 0.0f, I, 0.0f);
    }
}

// ---------------------------------------------------------------------------
// Kernel 1: WMMA partial sums. Grid = (32 ray-tiles, 16 segments) = 512 WGs.
// Each wave owns one 1024-dir chunk: 64x v_wmma_f32_16x16x4_f32 producing
// dot(normal_r, dir) for 16 rays x 16 dirs, clamp + intens-weighted
// accumulate, 4-step half-wave shfl reduction, chunk sums -> global (L2).
// ---------------------------------------------------------------------------
__global__ __launch_bounds__(256) void hdri_partial_kernel(
    const float* __restrict__ normal,    // [NPR,3]
    const float4* __restrict__ packed,   // pre-swizzled B + intens
    float* __restrict__ chunkSums)       // [NPR][NCHUNK]
{
    const int tid  = threadIdx.x;
    const int lane = tid & 31;
    const int wave = tid >> 5;                 // 0..7
    const int rayBase = blockIdx.x * RAYS_PER_WG;

    // ----- A matrix: rotated normals nr = (n.x, -n.z, -n.y) (angle = 0) -----
    v2f A;
    {
        const int m = lane & 15;
        const float* n = normal + (rayBase + m) * 3;
        const float nx = n[0];
        const float ny = -n[2];
        const float nz = -n[1];
        const bool lo = lane < 16;
        A.x = lo ? nx : nz;      // K=0 | K=2
        A.y = lo ? ny : 0.0f;    // K=1 | K=3 (pad)
    }

    // this wave's chunk
    const int chunk = blockIdx.y * 8 + wave;          // 0..127
    const int tileBase = chunk * (CHUNK / 16);        // 64 tiles per chunk
    const float4* pk = packed + (size_t)tileBase * 32 + lane;

    __builtin_prefetch(pk, 0, 1);                     // global_prefetch_b8

    v8f acc = {0.f, 0.f, 0.f, 0.f, 0.f, 0.f, 0.f, 0.f};

    for (int t = 0; t < CHUNK / 16; ++t) {
        const float4 rec = pk[t * 32];                // coalesced, stride 512B
        v2f B;
        B.x = rec.x;
        B.y = rec.y;
        const float w = rec.z;

        v8f c = {0.f, 0.f, 0.f, 0.f, 0.f, 0.f, 0.f, 0.f};
        v8f d = __builtin_amdgcn_wmma_f32_16x16x4_f32(
            false, A, false, B, (short)0, c, false, false);

#pragma unroll
        for (int r = 0; r < 8; ++r) {
            float m = d[r];
            m = fminf(fmaxf(m, 0.0f), 1.0f);          // v_max clamp
            acc[r] += m * w;                          // v_pk_fma_f32 pairs
        }
    }

    // reduce within 16-lane halves: lanes 0-15 -> ray r, 16-31 -> ray r+8
#pragma unroll
    for (int r = 0; r < 8; ++r) {
        float v = acc[r];
        v += __shfl_xor(v, 1, 32);
        v += __shfl_xor(v, 2, 32);
        v += __shfl_xor(v, 4, 32);
        v += __shfl_xor(v, 8, 32);
        if (lane == 0)  chunkSums[(rayBase + r)     * NCHUNK + chunk] = v;
        if (lane == 16) chunkSums[(rayBase + r + 8) * NCHUNK + chunk] = v;
    }
}

// ---------------------------------------------------------------------------
// Kernel 2: inverse-CDF sampling. Grid = 512 WGs (one ray), 8 waves = 8
// samples. Coarse scan over LDS chunk sums, then wave-parallel fine scan:
// 32-wide strips with shfl_up prefix sums + ballot first-exceed.
// ---------------------------------------------------------------------------
__global__ __launch_bounds__(256) void hdri_sample_kernel(
    const float* __restrict__ normal,
    const float4* __restrict__ packed,
    const float* __restrict__ chunkSums,
    const float* __restrict__ ps,
    float* __restrict__ out)
{
    __shared__ float cs[NCHUNK];

    const int ray  = blockIdx.x;
    const int tid  = threadIdx.x;
    const int lane = tid & 31;
    const int smp  = tid >> 5;                        // wave = sample 0..7

    if (tid < NCHUNK) cs[tid] = chunkSums[ray * NCHUNK + tid];
    __syncthreads();

    // total (identical computation in every wave -> identical float)
    float part = cs[lane] + cs[lane + 32] + cs[lane + 64] + cs[lane + 96];
#pragma unroll
    for (int m = 16; m >= 1; m >>= 1) part += __shfl_xor(part, m, 32);
    const float S = part;

    const float* n = normal + ray * 3;
    const float nx = n[0];
    const float ny = -n[2];
    const float nz = -n[1];

    const float target = ps[ray * NSR + smp] * S;

    // coarse: first chunk whose cumulative sum exceeds target (wave-uniform)
    float run = 0.0f;
    int c = 0;
    for (; c < NCHUNK - 1; ++c) {
        const float v = cs[c];
        if (run + v > target) break;
        run += v;
    }

    // fine: wave-parallel scan of the 1024-dir chunk in 32-wide strips
    const int nbase = c * CHUNK;
    int idx = HDRI_HW - 1;                            // fallback
    bool found = false;
    for (int strip = 0; strip < CHUNK / 32 && !found; ++strip) {
        const int nn   = nbase + strip * 32 + lane;
        const int tile = nn >> 4;
        const int slot = nn & 15;
        const float4 r0 = packed[tile * 32 + slot];        // x, y, I
        const float4 r1 = packed[tile * 32 + 16 + slot];   // z
        float m = nx * r0.x + ny * r0.y + nz * r1.x;
        m = fminf(fmaxf(m, 0.0f), 1.0f);
        const float v = r0.z * m;

        // inclusive prefix sum within the wave
        float incl = v;
#pragma unroll
        for (int off = 1; off < 32; off <<= 1) {
            const float t2 = __shfl_up(incl, off, 32);
            if (lane >= off) incl += t2;
        }
        const float stripTot = __shfl(incl, 31, 32);

        const unsigned long long ball = __ballot(run + incl > target);
        if (ball != 0ull) {
            const int first = __ffsll(ball) - 1;
            idx = nbase + strip * 32 + first;
            found = true;
        }
        run += stripTot;
    }
    if (idx > HDRI_HW - 1) idx = HDRI_HW - 1;

    // outputs (idx, S are wave-uniform)
    if (lane == 0) {
        const int tile = idx >> 4;
        const int slot = idx & 15;
        const float4 r0 = packed[tile * 32 + slot];
        const float4 r1 = packed[tile * 32 + 16 + slot];
        float m = nx * r0.x + ny * r0.y + nz * r1.x;
        m = fminf(fmaxf(m, 0.0f), 1.0f);
        const float val   = r0.z * m;
        const float normp = val / S;
        const float qx    = normp + 1e-7f;
        const float mpx   = (2.0f / (float)HDRI_HW) * m;
        const float ratio = mpx / qx;

        const int V = idx / HDRI_W;
        const int U = idx % HDRI_W;
        const int o = ray * NSR + smp;
        out[o]                 = (float)V;
        out[NPR * NSR + o]     = (float)U;
        out[2 * NPR * NSR + o] = ratio;
    }
}

// ---------------------------------------------------------------------------
extern "C" void kernel_launch(void* const* d_in, const int* in_sizes, int n_in,
                              void* d_out, int out_size, void* d_ws, size_t ws_size,
                              hipStream_t stream) {
    const float* normal = (const float*)d_in[0];   // [512,3]
    const float* envs   = (const float*)d_in[1];   // [1,256,512,3]
    const float* dirs   = (const float*)d_in[2];   // [256,512,3]
    const float* ps     = (const float*)d_in[3];   // [512,8,1]
    // d_in[4] = env_idxs: NUM_ENVS == 1 -> always 0, unused.
    (void)in_sizes; (void)n_in; (void)out_size; (void)ws_size;

    float4* packed    = (float4*)d_ws;                       // 4 MB
    float*  chunkSums = (float*)((char*)d_ws + (size_t)(HDRI_HW / 16) * 32 * 16);
    float*  out       = (float*)d_out;                       // Vs | Us | ratio

    hdri_pack_kernel<<<HDRI_HW / 256, 256, 0, stream>>>(envs, dirs, packed);

    dim3 gridP(NPR / RAYS_PER_WG, NSEG);                     // 32 x 16 = 512 WGs
    hdri_partial_kernel<<<gridP, 256, 0, stream>>>(normal, packed, chunkSums);

    hdri_sample_kernel<<<NPR, 256, 0, stream>>>(             // 512 WGs
        normal, packed, chunkSums, ps, out);
}